// ChamferDistanceLoss_65206193488018
// MI455X (gfx1250) — compile-verified
//
#include <hip/hip_runtime.h>

typedef __attribute__((ext_vector_type(2))) float v2f;
typedef __attribute__((ext_vector_type(8))) float v8f;

#define WAVES_PER_BLOCK 8
#define ROW_TILES_PER_WAVE 4   // 4 x 16 = 64 rows per wave

__global__ void chamfer_zero_out(float* out) { out[0] = 0.0f; }

// Single-instruction 3-way min (avoids fminf's canonicalize max(x,x) spam).
static __device__ __forceinline__ float min3f(float a, float b, float c) {
  float d;
  asm("v_min3_num_f32 %0, %1, %2, %3" : "=v"(d) : "v"(a), "v"(b), "v"(c));
  return d;
}

// Each wave owns R=4 consecutive 16-row tiles of `rows` and scans all columns
// of `cols`.  d[n,m] = ||r_n||^2 + ||c_m||^2 - 2 r.c  computed as
//   WMMA( A=[x,y,z,1] , B=[-2x,-2y,-2z,||c||^2] )  + ||r_n||^2 (added post-min)
__global__ __launch_bounds__(256) void chamfer_pass_kernel(
    const float* __restrict__ rows,   // (nB, nR, 3)
    const float* __restrict__ cols,   // (nB, nC, 3)
    int nB, int nR, int nC, float scale,
    float* __restrict__ out)
{
  const int lane = threadIdx.x & 31;
  const int wave = threadIdx.x >> 5;
  const int tilesPerBatch = nR >> 4;
  const int nTiles = nB * tilesPerBatch;
  const int tile0 = (blockIdx.x * WAVES_PER_BLOCK + wave) * ROW_TILES_PER_WAVE;
  if (tile0 >= nTiles) return;                      // uniform per wave

  const int b = tile0 / tilesPerBatch;              // R tiles never straddle a batch
  const bool hi = lane >= 16;
  const int  lj = lane & 15;

  // ---- A operands: 16x4 f32.  lanes 0-15: K0,K1 ; lanes 16-31: K2,K3 ----
  v2f   A[ROW_TILES_PER_WAVE];
  float rsq[ROW_TILES_PER_WAVE];                    // ||r||^2 for this lane's row
#pragma unroll
  for (int r = 0; r < ROW_TILES_PER_WAVE; ++r) {
    const int row = ((tile0 + r) % tilesPerBatch) * 16 + lj;
    const float* rp = rows + ((size_t)b * nR + (size_t)row) * 3;
    const float rx = rp[0], ry = rp[1], rz = rp[2];
    rsq[r] = rx * rx + ry * ry + rz * rz;
    A[r].x = hi ? rz : rx;        // K=2 : K=0
    A[r].y = hi ? 1.0f : ry;      // K=3 : K=1
  }

  v8f vmin[ROW_TILES_PER_WAVE];
#pragma unroll
  for (int r = 0; r < ROW_TILES_PER_WAVE; ++r)
#pragma unroll
    for (int i = 0; i < 8; ++i) vmin[r][i] = 3.0e38f;
  v8f zeroC = {};

  const float* cbase = cols + (size_t)b * nC * 3;
  const int nColTiles = nC >> 4;                    // 512: even, no tail

  for (int ct = 0; ct < nColTiles; ct += 2) {
    // ---- Two B operands (col tiles ct, ct+1): 4x16 f32 ----
    const float* cp0 = cbase + (size_t)(((ct + 0) << 4) + lj) * 3;
    const float* cp1 = cbase + (size_t)(((ct + 1) << 4) + lj) * 3;
    const float c0x = cp0[0], c0y = cp0[1], c0z = cp0[2];
    const float c1x = cp1[0], c1y = cp1[1], c1z = cp1[2];
    const float c0sq = c0x * c0x + c0y * c0y + c0z * c0z;
    const float c1sq = c1x * c1x + c1y * c1y + c1z * c1z;
    v2f B0, B1;
    B0.x = hi ? (-2.0f * c0z) : (-2.0f * c0x);   // K=2 : K=0
    B0.y = hi ? c0sq          : (-2.0f * c0y);   // K=3 : K=1
    B1.x = hi ? (-2.0f * c1z) : (-2.0f * c1x);
    B1.y = hi ? c1sq          : (-2.0f * c1y);

    // 8 independent WMMAs; pair outputs so each accumulator update is one min3
#pragma unroll
    for (int r = 0; r < ROW_TILES_PER_WAVE; ++r) {
      v8f c0 = __builtin_amdgcn_wmma_f32_16x16x4_f32(
          false, A[r], false, B0, (short)0, zeroC, false, false);
      v8f c1 = __builtin_amdgcn_wmma_f32_16x16x4_f32(
          false, A[r], false, B1, (short)0, zeroC, false, false);
#pragma unroll
      for (int v = 0; v < 8; ++v)
        vmin[r][v] = min3f(vmin[r][v], c0[v], c1[v]);
    }
  }

  // ---- Reduce: C/D layout puts row (v + 8*hi) in VGPR v across 16 lanes ----
#pragma unroll
  for (int r = 0; r < ROW_TILES_PER_WAVE; ++r) {
#pragma unroll
    for (int v = 0; v < 8; ++v) {
      float m = vmin[r][v];
      m = fminf(m, __shfl_xor(m, 1, 16));
      m = fminf(m, __shfl_xor(m, 2, 16));
      m = fminf(m, __shfl_xor(m, 4, 16));
      m = fminf(m, __shfl_xor(m, 8, 16));
      // ||r||^2 of this VGPR's row lives at segment-lane v (low) / v+8 (high)
      const int src = v + (hi ? 8 : 0);
      const float sq = __shfl(rsq[r], src, 16);
      if (lj == 0) atomicAdd(out, (m + sq) * scale);
    }
  }
}

extern "C" void kernel_launch(void* const* d_in, const int* in_sizes, int n_in,
                              void* d_out, int out_size, void* d_ws, size_t ws_size,
                              hipStream_t stream) {
  (void)n_in; (void)out_size; (void)d_ws; (void)ws_size;
  const float* pred = (const float*)d_in[0];
  const float* targ = (const float*)d_in[1];
  float* out = (float*)d_out;

  const int B = 4;
  const int N = in_sizes[0] / (3 * B);
  const int M = in_sizes[1] / (3 * B);

  chamfer_zero_out<<<1, 1, 0, stream>>>(out);

  // Pass 1: pred rows vs target cols  -> mean(dist1)
  {
    int tiles = B * (N / 16);
    int waves = tiles / ROW_TILES_PER_WAVE;
    int blocks = (waves + WAVES_PER_BLOCK - 1) / WAVES_PER_BLOCK;
    chamfer_pass_kernel<<<blocks, 32 * WAVES_PER_BLOCK, 0, stream>>>(
        pred, targ, B, N, M, 1.0f / (float)(B * N), out);
  }
  // Pass 2: target rows vs pred cols  -> mean(dist2)
  {
    int tiles = B * (M / 16);
    int waves = tiles / ROW_TILES_PER_WAVE;
    int blocks = (waves + WAVES_PER_BLOCK - 1) / WAVES_PER_BLOCK;
    chamfer_pass_kernel<<<blocks, 32 * WAVES_PER_BLOCK, 0, stream>>>(
        targ, pred, B, M, N, 1.0f / (float)(B * M), out);
  }
}